// ExRestSelfAtten_9543417331959
// MI455X (gfx1250) — compile-verified
//
#include <hip/hip_runtime.h>

typedef __attribute__((ext_vector_type(2))) float v2f;
typedef __attribute__((ext_vector_type(8))) float v8f;

#define BATCH 32768
#define SEQ   11
#define DIM   128
#define HID   32
#define OUTD  2
#define BATCH_PER_BLOCK 16
#define ROWS_PER_BLOCK  (BATCH_PER_BLOCK * SEQ)   // 176 rows, 11 wave32 tiles of 16
#define THREADS (ROWS_PER_BLOCK * OUTD)           // 352 = 11 waves

// Fused: h = relu(X@W1 + b1) + pos ; out = (window-sum h) @ (Wv@Wo) + bo
__global__ __launch_bounds__(THREADS)
void ExRestSelfAtten_fused_kernel(const float* __restrict__ x,
                                  const float* __restrict__ W1,
                                  const float* __restrict__ b1,
                                  const float* __restrict__ Wv,
                                  const float* __restrict__ pos,
                                  const float* __restrict__ Wo,
                                  const float* __restrict__ bo,
                                  float* __restrict__ out) {
  // Pre-swizzled WMMA B-fragments of W1: [kstep][ntile][lane] -> float2{W1[kk][n], W1[kk+1][n]}
  __shared__ v2f   Bfrag[32][2][32];               // 16 KB
  __shared__ float hbuf[ROWS_PER_BLOCK][HID + 1];  // padded stride vs bank conflicts
  __shared__ float wvo[HID][OUTD];
  __shared__ float gbuf[ROWS_PER_BLOCK][OUTD];

  const int tid = threadIdx.x;

  // ---- stage W1 fragments into LDS ----
  for (int idx = tid; idx < 32 * 2 * 32; idx += THREADS) {
    const int ks = idx >> 6;
    const int nt = (idx >> 5) & 1;
    const int ln = idx & 31;
    const int kk = ks * 4 + ((ln >> 4) << 1);      // lane-half selects K pair
    const int n  = nt * 16 + (ln & 15);            // lane%16 = column
    v2f v;
    v.x = W1[kk * HID + n];
    v.y = W1[(kk + 1) * HID + n];
    Bfrag[ks][nt][ln] = v;
  }
  // ---- Wvo = Wv @ Wo  (32x2, folded projection) ----
  if (tid < HID * OUTD) {
    const int hi = tid >> 1, o = tid & 1;
    float acc = 0.f;
    for (int k = 0; k < HID; ++k) acc += Wv[hi * HID + k] * Wo[k * OUTD + o];
    wvo[hi][o] = acc;
  }
  __syncthreads();

  // ---- GEMM: 16-row tile per wave, K=128 in 32 steps of 4, N=32 as two tiles ----
  const int lane  = tid & 31;
  const int wave  = tid >> 5;                      // 0..10
  const int rsel  = lane & 15;                     // A row within tile
  const int khalf = (lane >> 4) << 1;              // K sub-offset per lane-half
  const size_t m0 = (size_t)blockIdx.x * ROWS_PER_BLOCK + (size_t)wave * 16;
  const float* xrow = x + (m0 + rsel) * DIM + khalf;

  v8f c0 = {};
  v8f c1 = {};
#pragma unroll
  for (int ks = 0; ks < 32; ++ks) {
    v2f a   = *(const v2f*)(xrow + ks * 4);
    v2f bb0 = Bfrag[ks][0][lane];
    v2f bb1 = Bfrag[ks][1][lane];
    c0 = __builtin_amdgcn_wmma_f32_16x16x4_f32(false, a, false, bb0, (short)0, c0, false, false);
    c1 = __builtin_amdgcn_wmma_f32_16x16x4_f32(false, a, false, bb1, (short)0, c1, false, false);
  }

  // ---- epilogue: relu(acc + b1) + pos -> LDS ----
  // C/D layout: VGPR r: lanes0-15 -> M=r, N=lane ; lanes16-31 -> M=r+8, N=lane-16
  const int n0   = lane & 15;
  const int rofs = (lane >> 4) << 3;
#pragma unroll
  for (int r = 0; r < 8; ++r) {
    const int rl = wave * 16 + r + rofs;           // row within block (global row % 11 == rl % 11)
    const int t  = rl % SEQ;
    const float h0 = fmaxf(c0[r] + b1[n0], 0.f)      + pos[t * HID + n0];
    const float h1 = fmaxf(c1[r] + b1[n0 + 16], 0.f) + pos[t * HID + n0 + 16];
    hbuf[rl][n0]      = h0;
    hbuf[rl][n0 + 16] = h1;
  }
  __syncthreads();

  // ---- g[row][o] = h[row] . Wvo[:,o] ----
  {
    const int pair = tid >> 1, o = tid & 1;        // pair 0..175
    float acc = 0.f;
#pragma unroll
    for (int k = 0; k < HID; ++k) acc += hbuf[pair][k] * wvo[k][o];
    gbuf[pair][o] = acc;
  }
  __syncthreads();

  // ---- out[b,t,o] = bo[o] + sum_{j=max(0,t-5)..min(10,t+5)} g[b,j,o] ----
  {
    const int pair = tid >> 1, o = tid & 1;
    const int bl = pair / SEQ;
    const int t  = pair - bl * SEQ;
    const int lo = (t - 5 < 0) ? 0 : t - 5;
    const int hi = (t + 5 > SEQ - 1) ? SEQ - 1 : t + 5;
    float acc = bo[o];
    for (int j = lo; j <= hi; ++j) acc += gbuf[bl * SEQ + j][o];
    const size_t gb = (size_t)blockIdx.x * BATCH_PER_BLOCK + bl;
    out[(gb * SEQ + t) * OUTD + o] = acc;
  }
}

// attn_w = softmax over a singleton axis == all ones.
__global__ void ExRestSelfAtten_ones_kernel(float4* __restrict__ p) {
  const size_t i = (size_t)blockIdx.x * blockDim.x + threadIdx.x;
  p[i] = make_float4(1.f, 1.f, 1.f, 1.f);
}

extern "C" void kernel_launch(void* const* d_in, const int* in_sizes, int n_in,
                              void* d_out, int out_size, void* d_ws, size_t ws_size,
                              hipStream_t stream) {
  const float* x   = (const float*)d_in[0];  // (32768,11,128)
  const float* W1  = (const float*)d_in[1];  // (128,32)
  const float* b1  = (const float*)d_in[2];  // (1,1,32)
  // d_in[3] = Wq, d_in[4] = Wk : dead (softmax over singleton axis)
  const float* Wv  = (const float*)d_in[5];  // (32,32)
  const float* pos = (const float*)d_in[6];  // (1,11,32)
  const float* Wo  = (const float*)d_in[7];  // (32,2)
  const float* bo  = (const float*)d_in[8];  // (1,1,2)
  float* out = (float*)d_out;

  // out region: B*S*O = 720896 floats; attn_w region: B*S*WIN = 3964928 floats (all ones)
  ExRestSelfAtten_fused_kernel<<<BATCH / BATCH_PER_BLOCK, THREADS, 0, stream>>>(
      x, W1, b1, Wv, pos, Wo, bo, out);

  float4* attn = (float4*)(out + (size_t)BATCH * SEQ * OUTD);
  // 3964928 / 4 = 991232 float4 = 3872 blocks * 256 threads exactly
  ExRestSelfAtten_ones_kernel<<<3872, 256, 0, stream>>>(attn);
}